// DenseCRFLoss_19920058319365
// MI455X (gfx1250) — compile-verified
//
#include <hip/hip_runtime.h>
#include <hip/hip_bf16.h>

typedef __attribute__((ext_vector_type(2))) float v2f;
typedef __attribute__((ext_vector_type(8))) float v8f;

#define NIMG 4
#define KSEG 21
#define KPAD 24
#define FPAD 8
#define OH 64
#define OW 64
#define M 4096              // OH*OW
#define NTILES 256          // M/16

// ---------------------------------------------------------------------------
// Kernel 1: fused downsample + feature build into padded planes.
//   segw : [N][KPAD][M]  bilinear(2x2 avg) * nearest-ROI, channels 21..23 = 0
//   featw: [N][FPAD][M]  rows {x/50, y/50, r/.15, g/.15, b/.15, 0,0,0}
//   sqw  : [N][M]        sum of squared features
// ---------------------------------------------------------------------------
__global__ void crf_prep(const float* __restrict__ img,
                         const float* __restrict__ seg,
                         const float* __restrict__ roi,
                         float* __restrict__ segw,
                         float* __restrict__ featw,
                         float* __restrict__ sqw,
                         double* __restrict__ accum) {
    int idx = blockIdx.x * blockDim.x + threadIdx.x;
    if (idx == 0) *accum = 0.0;            // re-init every launch (graph replay safe)
    if (idx >= NIMG * M) return;
    int n = idx >> 12, p = idx & (M - 1);
    int y = p >> 6,    x = p & 63;
    int iy = y << 1,   ix = x << 1;        // nearest: floor(dst * 2)

    float r = roi[n * (128 * 128) + iy * 128 + ix];

    float f[5];
    f[0] = (float)x * (1.0f / 50.0f);      // sxy = SIGMA_XY * SCALE = 50
    f[1] = (float)y * (1.0f / 50.0f);
#pragma unroll
    for (int c = 0; c < 3; ++c)
        f[2 + c] = img[((n * 3 + c) * 128 + iy) * 128 + ix] * (1.0f / 0.15f);

    float sq = 0.0f;
#pragma unroll
    for (int k = 0; k < 5; ++k) sq += f[k] * f[k];
    sqw[n * M + p] = sq;

#pragma unroll
    for (int k = 0; k < FPAD; ++k)
        featw[(n * FPAD + k) * M + p] = (k < 5) ? f[k] : 0.0f;

    for (int k = 0; k < KPAD; ++k) {
        float s = 0.0f;
        if (k < KSEG) {
            const float* sp = seg + ((n * KSEG + k) * 128 + iy) * 128 + ix;
            // bilinear, align_corners=False, scale 0.5 -> exact 2x2 average
            s = 0.25f * (sp[0] + sp[1] + sp[128] + sp[129]) * r;
        }
        segw[(n * KPAD + k) * M + p] = s;
    }
}

// ---------------------------------------------------------------------------
// Kernel 2: pairwise tile sweep with V_WMMA_F32_16X16X4_F32.
// Grid: NIMG*NTILES workgroups of 256 threads (8 waves). Wave w owns i-tile of
// its WG and j-tiles {w, w+8, ...}. Per 16x16 tile-pair:
//   FF = feat_iT * feat_j   (2 WMMAs, K padded 5->8)
//   G  = seg_iT  * seg_j    (6 WMMAs, K padded 21->24)
//   acc += exp(-0.5*max(sq_i+sq_j-2FF,0)) * G
// ---------------------------------------------------------------------------
__global__ void __launch_bounds__(256)
crf_pairwise(const float* __restrict__ segw,
             const float* __restrict__ featw,
             const float* __restrict__ sqw,
             double* __restrict__ accum) {
    const int lane = threadIdx.x & 31;
    const int wv   = threadIdx.x >> 5;
    const int half = lane >> 4;       // lane group 0-15 / 16-31
    const int lm   = lane & 15;

    const int n  = blockIdx.x >> 8;
    const int it = blockIdx.x & (NTILES - 1);
    const int i0 = it << 4;

    const float* segn  = segw  + n * KPAD * M;
    const float* featn = featw + n * FPAD * M;
    const float* sqn   = sqw   + n * M;

    // A-matrix striping (16x4 f32): lane L, VGPR v -> A[L%16][v + 2*(L/16)]
    v2f aF[2], aS[6];
#pragma unroll
    for (int c = 0; c < 2; ++c) {
        aF[c].x = featn[(4 * c + 0 + 2 * half) * M + i0 + lm];
        aF[c].y = featn[(4 * c + 1 + 2 * half) * M + i0 + lm];
    }
#pragma unroll
    for (int c = 0; c < 6; ++c) {
        aS[c].x = segn[(4 * c + 0 + 2 * half) * M + i0 + lm];
        aS[c].y = segn[(4 * c + 1 + 2 * half) * M + i0 + lm];
    }
    // C/D striping: lane L, VGPR v -> D[v + 8*(L/16)][L%16]
    float sqi[8];
#pragma unroll
    for (int v = 0; v < 8; ++v) sqi[v] = sqn[i0 + v + 8 * half];

    float acc = 0.0f;
    for (int jt = wv; jt < NTILES; jt += 8) {   // uniform per wave: EXEC all 1s
        const int j0 = jt << 4;
        // B-matrix striping (4x16 f32): lane L, VGPR v -> B[v + 2*(L/16)][L%16]
        v2f bF[2], bS[6];
#pragma unroll
        for (int c = 0; c < 2; ++c) {
            bF[c].x = featn[(4 * c + 0 + 2 * half) * M + j0 + lm];
            bF[c].y = featn[(4 * c + 1 + 2 * half) * M + j0 + lm];
        }
#pragma unroll
        for (int c = 0; c < 6; ++c) {
            bS[c].x = segn[(4 * c + 0 + 2 * half) * M + j0 + lm];
            bS[c].y = segn[(4 * c + 1 + 2 * half) * M + j0 + lm];
        }
        const float sqj = sqn[j0 + lm];

        v8f FF = {};
        FF = __builtin_amdgcn_wmma_f32_16x16x4_f32(false, aF[0], false, bF[0],
                                                   (short)0, FF, false, false);
        FF = __builtin_amdgcn_wmma_f32_16x16x4_f32(false, aF[1], false, bF[1],
                                                   (short)0, FF, false, false);
        v8f G = {};
#pragma unroll
        for (int c = 0; c < 6; ++c)
            G = __builtin_amdgcn_wmma_f32_16x16x4_f32(false, aS[c], false, bS[c],
                                                      (short)0, G, false, false);
#pragma unroll
        for (int v = 0; v < 8; ++v) {
            float d2 = fmaxf(sqi[v] + sqj - 2.0f * FF[v], 0.0f);
            acc += __expf(-0.5f * d2) * G[v];
        }
    }

    // block reduction -> one f64 atomic per workgroup
    __shared__ float red[256];
    red[threadIdx.x] = acc;
    __syncthreads();
#pragma unroll
    for (int s = 128; s >= 32; s >>= 1) {
        if (threadIdx.x < s) red[threadIdx.x] += red[threadIdx.x + s];
        __syncthreads();
    }
    if (threadIdx.x == 0) {
        double s = 0.0;
        for (int i = 0; i < 32; ++i) s += (double)red[i];
        atomicAdd(accum, s);
    }
}

// ---------------------------------------------------------------------------
// Kernel 3: scalar epilogue.  loss = WEIGHT * (-sum / N)
// ---------------------------------------------------------------------------
__global__ void crf_final(const double* __restrict__ accum,
                          float* __restrict__ out) {
    out[0] = (float)(-2e-09 * (*accum) / (double)NIMG);
}

extern "C" void kernel_launch(void* const* d_in, const int* in_sizes, int n_in,
                              void* d_out, int out_size, void* d_ws, size_t ws_size,
                              hipStream_t stream) {
    const float* images = (const float*)d_in[0];   // [4,3,128,128]
    const float* segs   = (const float*)d_in[1];   // [4,21,128,128]
    const float* rois   = (const float*)d_in[2];   // [4,128,128]

    float* ws    = (float*)d_ws;
    float* segw  = ws;                             // 4*24*4096 floats
    float* featw = segw  + NIMG * KPAD * M;        // 4*8*4096 floats
    float* sqw   = featw + NIMG * FPAD * M;        // 4*4096 floats
    double* accum = (double*)(sqw + NIMG * M);     // 8-byte aligned (offset 2162688 B)

    crf_prep<<<(NIMG * M + 255) / 256, 256, 0, stream>>>(
        images, segs, rois, segw, featw, sqw, accum);

    crf_pairwise<<<NIMG * NTILES, 256, 0, stream>>>(segw, featw, sqw, accum);

    crf_final<<<1, 1, 0, stream>>>(accum, (float*)d_out);
}